// ConnectedFilterLayerWithImplicitJacobian_25984552141119
// MI455X (gfx1250) — compile-verified
//
#include <hip/hip_runtime.h>
#include <hip/hip_bf16.h>

typedef __attribute__((ext_vector_type(2))) float v2f;
typedef __attribute__((ext_vector_type(8))) float v8f;

// ---------------------------------------------------------------------------
// Phase 1+2+3: logits via V_WMMA_F32_16X16X4_F32.
// Weight lives in A row 0; attrs^T lives in B, so D[0][n] = logit(node n)
// arrives as VGPR0 on lane n (n=0..15): one logit per lane -> fully parallel
// sigmoid + Euler-tour scatter (every delta slot written exactly once, so
// plain stores, no atomics, no pre-zeroing).
// ---------------------------------------------------------------------------
__global__ void k_logits_scatter(const float* __restrict__ attrs,
                                 const float* __restrict__ weight,
                                 const float* __restrict__ bias,
                                 const float* __restrict__ residues,
                                 const int*   __restrict__ tpre,
                                 const int*   __restrict__ tpost,
                                 float*       __restrict__ delta,
                                 int N, int numTiles)
{
    const int lane        = threadIdx.x & 31;
    const int waveInBlock = threadIdx.x >> 5;
    const int tile        = blockIdx.x * (blockDim.x >> 5) + waveInBlock;
    if (tile >= numTiles) return;                 // wave-uniform exit: EXEC all-1s below

    const int  base   = tile * 16;
    const int  node   = base + (lane & 15);
    const bool hiHalf = (lane >= 16);

    // A-matrix (16x4 f32): lane m (0-15) holds {A[m][k],A[m][k+1]} for k-slab 0..1,
    // lane m+16 holds {A[m][k+2],A[m][k+3]}. Only row m==0 carries the weight.
    v2f a0 = {0.f, 0.f}, a1 = {0.f, 0.f};
    if ((lane & 15) == 0) {
        const int kk = hiHalf ? 2 : 0;
        a0.x = weight[kk + 0]; a0.y = weight[kk + 1];   // K slab 0..3
        a1.x = weight[kk + 4]; a1.y = weight[kk + 5];   // K slab 4..7
    }

    // B-matrix (4x16 f32): lane n (0-15): {B[0][n],B[1][n]} ; lane n+16: {B[2][n],B[3][n]}
    // with B[k][n] = attrs[node_n][k]  (i.e. attrs row transposed into a column).
    v2f b0 = {0.f, 0.f}, b1 = {0.f, 0.f};
    if (node < N) {
        const float* ap = attrs + (size_t)node * 8 + (hiHalf ? 2 : 0);
        b0.x = ap[0]; b0.y = ap[1];                     // K 0..3 half
        b1.x = ap[4]; b1.y = ap[5];                     // K 4..7 half
    }

    // D = A*B + C, fp32 accumulate over both K slabs (EXEC all-1s here)
    v8f c = {};
    c = __builtin_amdgcn_wmma_f32_16x16x4_f32(false, a0, false, b0, (short)0, c, false, false);
    c = __builtin_amdgcn_wmma_f32_16x16x4_f32(false, a1, false, b1, (short)0, c, false, false);

    // D row M=0 -> VGPR0 (c[0]) on lanes 0-15, N=lane. 16 nodes in parallel.
    if (!hiHalf && node < N) {
        const float logit = c[0] + bias[0];
        const float e     = __expf(-logit);
        const float sig   = __builtin_amdgcn_rcpf(1.0f + e);   // single v_rcp_f32
        const float f     = residues[node] * sig;
        delta[tpre[node]]  =  f;    // unique slot: plain store, no atomic
        delta[tpost[node]] = -f;    // unique slot
    }
}

// ---------------------------------------------------------------------------
// Phase 4: 3-kernel block scan over delta (T = 2N floats), in place.
// ---------------------------------------------------------------------------
#define SCAN_BLOCK 256
#define SCAN_ITEMS 8
#define SCAN_TILE  (SCAN_BLOCK * SCAN_ITEMS)   // 2048

__global__ void k_reduce_tiles(const float* __restrict__ delta,
                               float* __restrict__ partials, int T)
{
    __shared__ float lds[SCAN_BLOCK];
    const int base = blockIdx.x * SCAN_TILE + threadIdx.x * SCAN_ITEMS;
    float s = 0.f;
#pragma unroll
    for (int i = 0; i < SCAN_ITEMS; ++i) {
        const int idx = base + i;
        if (idx < T) s += delta[idx];
    }
    lds[threadIdx.x] = s;
    __syncthreads();
    for (int off = SCAN_BLOCK / 2; off > 0; off >>= 1) {
        if ((int)threadIdx.x < off) lds[threadIdx.x] += lds[threadIdx.x + off];
        __syncthreads();
    }
    if (threadIdx.x == 0) partials[blockIdx.x] = lds[0];
}

__global__ void k_scan_partials(float* __restrict__ partials, int n)
{
    __shared__ float lds[1024];
    __shared__ float carry_s;
    if (threadIdx.x == 0) carry_s = 0.f;
    __syncthreads();
    for (int start = 0; start < n; start += 1024) {
        const int i = start + (int)threadIdx.x;
        const float x = (i < n) ? partials[i] : 0.f;
        lds[threadIdx.x] = x;
        __syncthreads();
        for (int off = 1; off < 1024; off <<= 1) {
            const float t = ((int)threadIdx.x >= off) ? lds[threadIdx.x - off] : 0.f;
            __syncthreads();
            lds[threadIdx.x] += t;
            __syncthreads();
        }
        const float carry = carry_s;
        if (i < n) partials[i] = carry + lds[threadIdx.x] - x;  // exclusive prefix
        __syncthreads();
        if (threadIdx.x == 1023) carry_s = carry + lds[1023];
        __syncthreads();
    }
}

__global__ void k_scan_apply(float* __restrict__ delta,
                             const float* __restrict__ partials, int T)
{
    __shared__ float lds[SCAN_BLOCK];
    const int base = blockIdx.x * SCAN_TILE + threadIdx.x * SCAN_ITEMS;
    float v[SCAN_ITEMS];
    float s = 0.f;
#pragma unroll
    for (int i = 0; i < SCAN_ITEMS; ++i) {
        const int idx = base + i;
        const float x = (idx < T) ? delta[idx] : 0.f;
        s += x;
        v[i] = s;                                   // thread-local inclusive scan
    }
    lds[threadIdx.x] = s;
    __syncthreads();
    for (int off = 1; off < SCAN_BLOCK; off <<= 1) {
        const float t = ((int)threadIdx.x >= off) ? lds[threadIdx.x - off] : 0.f;
        __syncthreads();
        lds[threadIdx.x] += t;
        __syncthreads();
    }
    const float texcl = lds[threadIdx.x] - s + partials[blockIdx.x];
#pragma unroll
    for (int i = 0; i < SCAN_ITEMS; ++i) {
        const int idx = base + i;
        if (idx < T) delta[idx] = v[i] + texcl;     // in-place inclusive cumsum
    }
}

// ---------------------------------------------------------------------------
// Phase 5a: hoist inner indirection -> ynode[i] = cumsum[tpre[i]] (L2-resident)
// ---------------------------------------------------------------------------
__global__ void k_node_gather(const float* __restrict__ cumsum,
                              const int*   __restrict__ tpre,
                              float*       __restrict__ ynode, int N)
{
    const int i = blockIdx.x * blockDim.x + threadIdx.x;
    if (i < N) ynode[i] = cumsum[tpre[i]];
}

// ---------------------------------------------------------------------------
// Phase 5b: pixel gather, 4 pixels/thread (int4 load / float4 store)
// ---------------------------------------------------------------------------
__global__ void k_pixel_gather(const float* __restrict__ ynode,
                               const int*   __restrict__ nop,
                               float*       __restrict__ y, int P)
{
    const int i4 = (blockIdx.x * blockDim.x + threadIdx.x) * 4;
    if (i4 + 3 < P) {
        const int4 n = *reinterpret_cast<const int4*>(nop + i4);
        float4 o;
        o.x = ynode[n.x];
        o.y = ynode[n.y];
        o.z = ynode[n.z];
        o.w = ynode[n.w];
        *reinterpret_cast<float4*>(y + i4) = o;
    } else {
        for (int i = i4; i < P; ++i) y[i] = ynode[nop[i]];
    }
}

// ---------------------------------------------------------------------------
extern "C" void kernel_launch(void* const* d_in, const int* in_sizes, int n_in,
                              void* d_out, int out_size, void* d_ws, size_t ws_size,
                              hipStream_t stream)
{
    const float* weight   = (const float*)d_in[0];   // (K,)
    const float* bias     = (const float*)d_in[1];   // (1,)
    const float* residues = (const float*)d_in[2];   // (N,)
    const float* attrs    = (const float*)d_in[3];   // (N,K)
    const int*   tpre     = (const int*)d_in[4];     // (N,)
    const int*   tpost    = (const int*)d_in[5];     // (N,)
    const int*   nop      = (const int*)d_in[6];     // (P,)
    float*       y        = (float*)d_out;           // (P,)

    const int N = in_sizes[2];
    const int P = in_sizes[6];
    const int T = 2 * N;

    // workspace carve-up: delta/cumsum (T), partials, ynode (N)  (~6 MB total)
    float* delta         = (float*)d_ws;
    const int numScanBlk = (T + SCAN_TILE - 1) / SCAN_TILE;
    float* partials      = delta + T;
    float* ynode         = partials + ((numScanBlk + 255) & ~255);

    // phase 1-3: WMMA logits + sigmoid + scatter
    const int numTiles = (N + 15) / 16;
    const int wavesPerBlock = 8;                     // 256 threads = 8 wave32
    const int blocks1 = (numTiles + wavesPerBlock - 1) / wavesPerBlock;
    k_logits_scatter<<<blocks1, 256, 0, stream>>>(attrs, weight, bias, residues,
                                                  tpre, tpost, delta, N, numTiles);

    // phase 4: scan
    k_reduce_tiles<<<numScanBlk, SCAN_BLOCK, 0, stream>>>(delta, partials, T);
    k_scan_partials<<<1, 1024, 0, stream>>>(partials, numScanBlk);
    k_scan_apply<<<numScanBlk, SCAN_BLOCK, 0, stream>>>(delta, partials, T);

    // phase 5: gathers
    const int blocksN = (N + 255) / 256;
    k_node_gather<<<blocksN, 256, 0, stream>>>(delta, tpre, ynode, N);

    const int blocksP = ((P + 3) / 4 + 255) / 256;
    k_pixel_gather<<<blocksP, 256, 0, stream>>>(ynode, nop, y, P);
}